// LatticeGnn_62852551409829
// MI455X (gfx1250) — compile-verified
//
#include <hip/hip_runtime.h>
#include <stdint.h>

// ---------------------------------------------------------------------------
// LatticeGnn on MI455X (gfx1250).
// GCNConv with in=1/out=2 collapses to per-node scalars:
//   deg[i] = 1 + #{e: col[e]==i};  dinv = rsqrt(deg)
//   s[i]   = dinv[i]*x[i] + sum_{e: col[e]==i} dinv[row[e]]*x[row[e]]
//   g[i]   = relu(dinv*s*W0 + b0) + relu(dinv*s*W1 + b1)
//   out[k] = sigmoid(0.5*(g[row[k]]+g[col[k]]+g[row[k+H]]+g[col[k+H]]))
// Memory-bound; CDNA5 path: async global->LDS streaming of edge indices
// (global_load_async_to_lds_b128 / s_wait_asynccnt), exact integer degree
// atomics, NT hints on the single-use streams.
// ---------------------------------------------------------------------------

typedef int   v4i __attribute__((ext_vector_type(4)));   // native vectors for
typedef float v4f __attribute__((ext_vector_type(4)));   // nontemporal builtins

__device__ __forceinline__ void async_b128_to_lds(uint32_t lds_byte_addr,
                                                  const void* gptr) {
  // LDS dest byte-offset in a VGPR, 64-bit global address in a VGPR pair.
  // Tracked by ASYNCcnt.
  asm volatile("global_load_async_to_lds_b128 %0, %1, off"
               :
               : "v"(lds_byte_addr), "v"((uint64_t)(uintptr_t)gptr)
               : "memory");
}

__device__ __forceinline__ void wait_async0() {
  asm volatile("s_wait_asynccnt 0" ::: "memory");
}

__device__ __forceinline__ uint32_t lds_addr_of(const void* p) {
  // Generic pointer to __shared__: low 32 bits are the LDS byte offset.
  return (uint32_t)(uintptr_t)p;
}

// ---- Kernel 1: deg[i] = 1 (self-loop contribution), exact integer ---------
__global__ __launch_bounds__(256) void k_init_deg(unsigned* __restrict__ deg, int N) {
  int i = blockIdx.x * blockDim.x + threadIdx.x;
  if (i < N) deg[i] = 1u;
}

// ---- Kernel 2: deg[col[e]] += 1 (4 edges/lane, NT streamed, int atomics) --
__global__ __launch_bounds__(256) void k_count_deg(const int* __restrict__ col,
                                                   unsigned* __restrict__ deg, int e4) {
  int i = blockIdx.x * blockDim.x + threadIdx.x;
  if (i >= e4) return;
  v4i c = __builtin_nontemporal_load(((const v4i*)col) + i);
  atomicAdd(&deg[c.x], 1u);
  atomicAdd(&deg[c.y], 1u);
  atomicAdd(&deg[c.z], 1u);
  atomicAdd(&deg[c.w], 1u);
}

// ---- Kernel 3: dinv = rsqrt(deg); t = s = dinv * x  (deg >= 1 always) -----
__global__ __launch_bounds__(256) void k_node_prep(const float* __restrict__ x,
                                                   const unsigned* __restrict__ deg,
                                                   float* __restrict__ dinv,
                                                   float* __restrict__ t,
                                                   float* __restrict__ s, int N) {
  int i = blockIdx.x * blockDim.x + threadIdx.x;
  if (i >= N) return;
  float d = rsqrtf((float)deg[i]);
  dinv[i] = d;             // overwrites the deg buffer (same storage)
  float ti = d * x[i];
  t[i] = ti;               // gathered per edge in the message pass
  s[i] = ti;               // self-loop term pre-seeded
}

// ---- Kernel 4: s[col[e]] += t[row[e]]  (main scatter; async LDS stream) ---
__global__ __launch_bounds__(256) void k_scatter_msg(const int* __restrict__ row,
                                                     const int* __restrict__ col,
                                                     const float* __restrict__ t,
                                                     float* __restrict__ s, int e4) {
  __shared__ int4 lbuf[512];  // [0:256) row quads, [256:512) col quads
  int i = blockIdx.x * blockDim.x + threadIdx.x;
  uint32_t base = lds_addr_of(&lbuf[0]);
  if (i < e4) {
    async_b128_to_lds(base + threadIdx.x * 16u,           ((const int4*)row) + i);
    async_b128_to_lds(base + (256u + threadIdx.x) * 16u,  ((const int4*)col) + i);
  }
  wait_async0();
  if (i >= e4) return;
  int4 r = lbuf[threadIdx.x];
  int4 c = lbuf[256 + threadIdx.x];
  atomicAdd(&s[c.x], t[r.x]);
  atomicAdd(&s[c.y], t[r.y]);
  atomicAdd(&s[c.z], t[r.z]);
  atomicAdd(&s[c.w], t[r.w]);
}

// ---- Kernel 5: g[i] = relu(a*W0+b0) + relu(a*W1+b1), a = dinv*s -----------
__global__ __launch_bounds__(256) void k_node_g(const float* __restrict__ dinv,
                                                const float* __restrict__ s,
                                                const float* __restrict__ W,
                                                const float* __restrict__ b,
                                                float* __restrict__ g, int N) {
  int i = blockIdx.x * blockDim.x + threadIdx.x;
  if (i >= N) return;
  float a  = dinv[i] * s[i];
  float h0 = fmaxf(fmaf(a, W[0], b[0]), 0.0f);
  float h1 = fmaxf(fmaf(a, W[1], b[1]), 0.0f);
  g[i] = h0 + h1;
}

// ---- Kernel 6: out[k] = sigmoid(0.5*(g[r1]+g[c1]+g[r2]+g[c2])) ------------
__global__ __launch_bounds__(256) void k_edge_out(const int* __restrict__ row,
                                                  const int* __restrict__ col,
                                                  const float* __restrict__ g,
                                                  float* __restrict__ out,
                                                  int h4, int H) {
  __shared__ int4 lbuf[1024];  // 4 index quads per lane (16 KB)
  int i = blockIdx.x * blockDim.x + threadIdx.x;
  uint32_t base = lds_addr_of(&lbuf[0]);
  if (i < h4) {
    async_b128_to_lds(base + threadIdx.x * 16u,            ((const int4*)row) + i);
    async_b128_to_lds(base + (256u  + threadIdx.x) * 16u,  ((const int4*)col) + i);
    async_b128_to_lds(base + (512u  + threadIdx.x) * 16u,  ((const int4*)(row + H)) + i);
    async_b128_to_lds(base + (768u  + threadIdx.x) * 16u,  ((const int4*)(col + H)) + i);
  }
  wait_async0();
  if (i >= h4) return;
  int4 r1 = lbuf[threadIdx.x];
  int4 c1 = lbuf[256 + threadIdx.x];
  int4 r2 = lbuf[512 + threadIdx.x];
  int4 c2 = lbuf[768 + threadIdx.x];
  v4f o;
  float y;
  y = 0.5f * (g[r1.x] + g[c1.x] + g[r2.x] + g[c2.x]);
  o.x = 1.0f / (1.0f + __expf(-y));
  y = 0.5f * (g[r1.y] + g[c1.y] + g[r2.y] + g[c2.y]);
  o.y = 1.0f / (1.0f + __expf(-y));
  y = 0.5f * (g[r1.z] + g[c1.z] + g[r2.z] + g[c2.z]);
  o.z = 1.0f / (1.0f + __expf(-y));
  y = 0.5f * (g[r1.w] + g[c1.w] + g[r2.w] + g[c2.w]);
  o.w = 1.0f / (1.0f + __expf(-y));
  __builtin_nontemporal_store(o, ((v4f*)out) + i);  // single-use 17 MB stream
}

// ---------------------------------------------------------------------------
extern "C" void kernel_launch(void* const* d_in, const int* in_sizes, int n_in,
                              void* d_out, int out_size, void* d_ws, size_t ws_size,
                              hipStream_t stream) {
  const float* x  = (const float*)d_in[0];   // [N,1]
  const float* W  = (const float*)d_in[1];   // [1,2]
  const float* bb = (const float*)d_in[2];   // [2]
  const int*   ei = (const int*)d_in[3];     // [2,E] (int32 per harness contract)

  const int N = in_sizes[0];                 // 132651
  const int E = in_sizes[3] / 2;             // 8489664
  const int H = E / 2;                       // 4244832 == out_size
  const int e4 = E / 4;                      // E divisible by 4
  const int h4 = H / 4;                      // H divisible by 4

  const int* row = ei;
  const int* col = ei + E;

  // workspace: deg(uint)/dinv(float, same storage) | t | s | g  (~2.1 MB)
  unsigned* deg = (unsigned*)d_ws;
  float* dinv = (float*)d_ws;
  float* t    = dinv + N;
  float* s    = t + N;
  float* g    = s + N;

  const dim3 B(256);
  const int gridN = (N  + 255) / 256;
  const int gridE = (e4 + 255) / 256;
  const int gridH = (h4 + 255) / 256;

  k_init_deg   <<<gridN, B, 0, stream>>>(deg, N);
  k_count_deg  <<<gridE, B, 0, stream>>>(col, deg, e4);
  k_node_prep  <<<gridN, B, 0, stream>>>(x, deg, dinv, t, s, N);
  k_scatter_msg<<<gridE, B, 0, stream>>>(row, col, t, s, e4);
  k_node_g     <<<gridN, B, 0, stream>>>(dinv, s, W, bb, g, N);
  k_edge_out   <<<gridH, B, 0, stream>>>(row, col, g, (float*)d_out, h4, H);
}